// ReadoutLayer_53446573031872
// MI455X (gfx1250) — compile-verified
//
#include <hip/hip_runtime.h>

// Masked sum-pool readout: out[b,d] = sum_n H[b,n,d] * (op[b,n] != 5)
// B=4096, N=64, D=256. Memory-bound (~0.5 flop/byte); accumulation is routed
// through V_WMMA_F32_16X16X4_F32 with a layout-invariant all-ones B operand:
//   D[m,n] = sum_k A[m,k] * 1,  A[m,k] = mask[n0+k] * H[b, n0+k, dbase+m]
// accumulated over n0 via the C operand.

typedef __attribute__((ext_vector_type(2))) float v2f;
typedef __attribute__((ext_vector_type(8))) float v8f;

#define N_NODES 64
#define D_FEAT  256
#define SKIP_OP 5

__global__ __launch_bounds__(256) void readout_wmma_kernel(
    const float* __restrict__ H,      // [B, 64, 256] f32
    const int*   __restrict__ op,     // [B, 64] (harness passes integer inputs as int)
    float*       __restrict__ out)    // [B, 256] f32
{
    __shared__ float smask[N_NODES];

    const int b     = blockIdx.x;
    const int tid   = threadIdx.x;
    const int wave  = tid >> 5;        // 8 waves per block
    const int lane  = tid & 31;
    const int m     = lane & 15;       // M index of A / output feature offset
    const int khalf = lane >> 4;       // 0 -> K={0,1}, 1 -> K={2,3} (ISA A layout)

    // Convert mask to f32 once per block (64 values, reused by all 8 waves).
    if (tid < N_NODES) {
        const int o = op[b * N_NODES + tid];
        smask[tid] = (o != SKIP_OP) ? 1.0f : 0.0f;
    }
    __syncthreads();

    const float* Hb    = H + (size_t)b * (N_NODES * D_FEAT);
    const int    dbase = wave * 32;    // this wave reduces d in [dbase, dbase+32)

    v2f ones; ones.x = 1.0f; ones.y = 1.0f;   // B matrix: all ones (layout-proof)
    v8f c0 = {};                               // accumulates out[dbase .. dbase+15]
    v8f c1 = {};                               // accumulates out[dbase+16 .. dbase+31]

    #pragma unroll
    for (int nb = 0; nb < N_NODES; nb += 4) {
        const int   r   = nb + 2 * khalf;          // node row for this lane half
        const float mk0 = smask[r];
        const float mk1 = smask[r + 1];
        const float* row = Hb + (size_t)r * D_FEAT + dbase + m;

        v2f a0, a1;
        a0.x = __builtin_nontemporal_load(row)              * mk0; // K=2h,   d=dbase+m
        a0.y = __builtin_nontemporal_load(row + D_FEAT)     * mk1; // K=2h+1, d=dbase+m
        a1.x = __builtin_nontemporal_load(row + 16)         * mk0; // K=2h,   d=dbase+16+m
        a1.y = __builtin_nontemporal_load(row + 16 + D_FEAT)* mk1; // K=2h+1, d=dbase+16+m

        // 8 args: (neg_a, A, neg_b, B, c_mod, C, reuse_a, reuse_b)
        c0 = __builtin_amdgcn_wmma_f32_16x16x4_f32(false, a0, false, ones,
                                                   (short)0, c0, false, false);
        c1 = __builtin_amdgcn_wmma_f32_16x16x4_f32(false, a1, false, ones,
                                                   (short)0, c1, false, false);
    }

    // C/D layout (32-bit 16x16): VGPR r = M=r on lanes 0-15, M=r+8 on lanes 16-31,
    // value identical across N columns. Lane 0 writes d = dbase+0..7 (+16..23),
    // lane 16 writes d = dbase+8..15 (+24..31).
    if (m == 0) {
        float* ob = out + (size_t)b * D_FEAT + dbase + khalf * 8;
        #pragma unroll
        for (int r2 = 0; r2 < 8; ++r2) {
            __builtin_nontemporal_store(c0[r2], ob + r2);
            __builtin_nontemporal_store(c1[r2], ob + 16 + r2);
        }
    }
}

extern "C" void kernel_launch(void* const* d_in, const int* in_sizes, int n_in,
                              void* d_out, int out_size, void* d_ws, size_t ws_size,
                              hipStream_t stream) {
    (void)n_in; (void)out_size; (void)d_ws; (void)ws_size;
    const float* H  = (const float*)d_in[0];   // [B,64,256] f32
    const int*   op = (const int*)d_in[1];     // [B,64] integer
    float*       out = (float*)d_out;          // [B,256] f32

    const int B = in_sizes[1] / N_NODES;       // 4096 for the reference shapes
    readout_wmma_kernel<<<B, 256, 0, stream>>>(H, op, out);
}